// SmoothingLayer_32315333935230
// MI455X (gfx1250) — compile-verified
//
#include <hip/hip_runtime.h>

// Fused 3-axis separable Gaussian blur (sigma=1, 7 taps, SAME/zero pad) on
// (2,160,160,160,4) f32 NDHWC; C=4 -> one float4 per voxel.
//
// Z-streaming design: each workgroup owns a 16x16 xy tile and streams a
// 20-slice z-chunk. Per z-slice: async-DMA (ASYNCcnt) the 22x22 halo slice
// into double-buffered LDS (prefetch overlaps compute), x-blur + y-blur in
// LDS, z-blur entirely in a 7-deep register ring -> coalesced float4 stores.
// Min HBM traffic 262 MB -> ~11 us @ 23.3 TB/s; LDS 21.8 KB/WG for occupancy.

namespace {
constexpr int Sdim   = 160;
constexpr int TX     = 16;               // output tile x
constexpr int TY     = 16;               // output tile y
constexpr int HALO   = 3;
constexpr int IX     = TX + 2 * HALO;    // 22 input cols
constexpr int IY     = TY + 2 * HALO;    // 22 input rows
constexpr int ROWP   = IX + 1;           // 23: padded LDS row stride (bank decorrelation)
constexpr int CH     = 20;               // output z-slices per workgroup
constexpr int NSLICE = CH + 2 * HALO;    // 26 input slices streamed
constexpr int THREADS = 256;             // one thread per xy output column
constexpr int NT_XY  = Sdim / TX;        // 10
constexpr int NT_Z   = Sdim / CH;        // 8
}

// 128-bit int vector matching the async builtin's parameter type
typedef int v4i __attribute__((vector_size(4 * sizeof(int))));
typedef __attribute__((address_space(1))) v4i glb_v4i_t;
typedef __attribute__((address_space(3))) v4i lds_v4i_t;

__device__ __forceinline__ void async_copy_f4_to_lds(const float4* g, float4* l) {
#if defined(__gfx1250__) && __has_builtin(__builtin_amdgcn_global_load_async_to_lds_b128)
  __builtin_amdgcn_global_load_async_to_lds_b128(
      (glb_v4i_t*)const_cast<float4*>(g), (lds_v4i_t*)l, 0, 0);
#else
  *l = *g;   // host-pass / non-async fallback
#endif
}

__device__ __forceinline__ void wait_async_zero() {
#if defined(__gfx1250__) && __has_builtin(__builtin_amdgcn_s_wait_asynccnt)
  __builtin_amdgcn_s_wait_asynccnt(0);
#elif defined(__gfx1250__)
  asm volatile("s_wait_asynccnt 0" ::: "memory");
#endif
}

__global__ void __launch_bounds__(THREADS)
gauss3d_zstream_kernel(const float4* __restrict__ in, float4* __restrict__ out) {
  __shared__ float4 inS[2][IY * ROWP];   // double-buffered raw input slice
  __shared__ float4 xS[IY * TX];         // x-blurred slice (y:22, x:16)

  const int tid = threadIdx.x;
  const int x0  = blockIdx.x * TX;
  const int y0  = blockIdx.y * TY;
  const int n   = blockIdx.z / NT_Z;
  const int z0  = (blockIdx.z % NT_Z) * CH;

  const int lx = tid % TX;               // this thread's output column
  const int ly = tid / TX;

  // Normalized 7-tap Gaussian, sigma=1.
  const float w[7] = {0.00443305f, 0.05400558f, 0.24203623f, 0.39905028f,
                      0.24203623f, 0.05400558f, 0.00443305f};

  // Issue async loads (or zero-fills for SAME padding) for input slice s.
  auto issue_slice = [&](int s, int b) {
    const bool z_ok = (unsigned)s < (unsigned)Sdim;
    for (int i = tid; i < IY * IX; i += THREADS) {
      const int xin = i % IX;
      const int yin = i / IX;
      const int gx  = x0 + xin - HALO;
      const int gy  = y0 + yin - HALO;
      float4* lp = &inS[b][yin * ROWP + xin];
      if (z_ok && (unsigned)gx < (unsigned)Sdim && (unsigned)gy < (unsigned)Sdim) {
        const float4* gp = in + (((n * Sdim + s) * Sdim + gy) * Sdim + gx);
        async_copy_f4_to_lds(gp, lp);
      } else {
        *lp = make_float4(0.f, 0.f, 0.f, 0.f);
      }
    }
  };

  // Prologue: land the first slice.
  issue_slice(z0 - HALO, 0);
  wait_async_zero();
  __syncthreads();

  // 7-deep z accumulator ring in registers: acc[0] is oldest (completes next).
  float4 acc[7];
#pragma unroll
  for (int j = 0; j < 7; ++j) acc[j] = make_float4(0.f, 0.f, 0.f, 0.f);

  int cur = 0;
  for (int si = 0; si < NSLICE; ++si) {
    const int s   = z0 - HALO + si;
    const int nxt = cur ^ 1;

    // Prefetch next slice while we compute this one (ASYNCcnt pipelining).
    if (si + 1 < NSLICE) issue_slice(s + 1, nxt);

    // ---- x-blur: 2 outputs/thread via 8-wide sliding window (4 reads/out) ----
    if (tid < (IY * TX) / 2) {                 // 176 threads
      const int yr = tid >> 3;                 // 0..21
      const int xb = (tid & 7) * 2;            // 0,2,...,14
      const float4* row = &inS[cur][yr * ROWP + xb];
      float4 r[8];
#pragma unroll
      for (int k = 0; k < 8; ++k) r[k] = row[k];
      float o0x = 0.f, o0y = 0.f, o0z = 0.f, o0w = 0.f;
      float o1x = 0.f, o1y = 0.f, o1z = 0.f, o1w = 0.f;
#pragma unroll
      for (int k = 0; k < 7; ++k) {
        o0x = fmaf(w[k], r[k].x, o0x);
        o0y = fmaf(w[k], r[k].y, o0y);
        o0z = fmaf(w[k], r[k].z, o0z);
        o0w = fmaf(w[k], r[k].w, o0w);
        o1x = fmaf(w[k], r[k + 1].x, o1x);
        o1y = fmaf(w[k], r[k + 1].y, o1y);
        o1z = fmaf(w[k], r[k + 1].z, o1z);
        o1w = fmaf(w[k], r[k + 1].w, o1w);
      }
      xS[yr * TX + xb]     = make_float4(o0x, o0y, o0z, o0w);
      xS[yr * TX + xb + 1] = make_float4(o1x, o1y, o1z, o1w);
    }
    __syncthreads();

    // ---- y-blur at this thread's (lx, ly) ----
    float vx = 0.f, vy = 0.f, vz = 0.f, vw = 0.f;
#pragma unroll
    for (int k = 0; k < 7; ++k) {
      const float4 u = xS[(ly + k) * TX + lx];
      vx = fmaf(w[k], u.x, vx);
      vy = fmaf(w[k], u.y, vy);
      vz = fmaf(w[k], u.z, vz);
      vw = fmaf(w[k], u.w, vw);
    }

    // ---- z-blur: fold slice s into the register ring ----
#pragma unroll
    for (int j = 0; j < 7; ++j) {
      acc[j].x = fmaf(w[6 - j], vx, acc[j].x);
      acc[j].y = fmaf(w[6 - j], vy, acc[j].y);
      acc[j].z = fmaf(w[6 - j], vz, acc[j].z);
      acc[j].w = fmaf(w[6 - j], vw, acc[j].w);
    }

    if (si >= 6) {
      const int zo = s - HALO;  // z0 + (si - 6), always in [z0, z0+CH)
      out[((n * Sdim + zo) * Sdim + (y0 + ly)) * Sdim + (x0 + lx)] = acc[0];
    }

    // Rotate the ring.
#pragma unroll
    for (int j = 0; j < 6; ++j) acc[j] = acc[j + 1];
    acc[6] = make_float4(0.f, 0.f, 0.f, 0.f);

    // Next slice must be resident (this wave's DMA) and all xS reads done
    // before the buffers are overwritten next iteration.
    wait_async_zero();
    __syncthreads();
    cur = nxt;
  }
}

extern "C" void kernel_launch(void* const* d_in, const int* in_sizes, int n_in,
                              void* d_out, int out_size, void* d_ws, size_t ws_size,
                              hipStream_t stream) {
  (void)in_sizes; (void)n_in; (void)out_size; (void)d_ws; (void)ws_size;
  const float4* in = (const float4*)d_in[0];
  float4* out      = (float4*)d_out;

  dim3 grid(NT_XY, NT_XY, 2 * NT_Z);   // 10 x 10 x 16 = 1600 workgroups
  dim3 block(THREADS);
  gauss3d_zstream_kernel<<<grid, block, 0, stream>>>(in, out);
}